// EnvCollLoss_9998683865516
// MI455X (gfx1250) — compile-verified
//
#include <hip/hip_runtime.h>
#include <math.h>

#define NAG   256
#define TT    100
#define NTOT  (NAG*TT)      // 25600 rows
#define RH    2048
#define RW    2048
#define PUU   10
#define PVV   20
#define NP    (PUU*PVV)     // 200 footprint points
#define NTILES 13           // ceil(200/16)
#define BIGF  1.0e30f
#define COLLT 1.0e29f

typedef __attribute__((ext_vector_type(2))) float v2f;
typedef __attribute__((ext_vector_type(8))) float v8f;

// One wave32 processes 16 consecutive flat rows x 208 footprint points.
// Three WMMA f32 16x16x4 per 16-point tile:
//   Dx = A *Bx + Cx  -> px*inv_dx   (A row = inv_dx*[L*hx, -W*hy, L*hy, W*hx])
//   Dy = A *By + Cy  -> py*inv_dx
//   Dd = A2*Bd + 0   -> d2 = L^2*uu^2 + W^2*vv^2  (rotation preserves norm)
// Collision min kept as bitcast integer min: inputs are non-negative finite
// floats, so v_min_i32 orders identically to v_min_num_f32 without the
// NaN-canonicalize ops clang wraps around fminf.
__global__ __launch_bounds__(128) void env_coll_kernel(
    const float* __restrict__ traj,     // [NTOT,4] cx,cy,hx,hy
    const float* __restrict__ veh_att,  // [NAG,2]  L,W
    const float* __restrict__ raster,   // [4,RH,RW]
    const int*   __restrict__ mapixes,  // [NAG]
    const float* __restrict__ dxp,      // [1]
    float*       __restrict__ out)      // [NTOT]
{
    const int lane = threadIdx.x & 31;
    const int wave = blockIdx.x * (blockDim.x >> 5) + (threadIdx.x >> 5);
    const int n0   = wave * 16;
    if (n0 >= NTOT) return;                     // wave-uniform; EXEC stays all-ones

    const float inv_dx = 1.0f / dxp[0];         // dx = 0.25 -> exact

    // ---- per-lane row data (lane l owns row n0 + l%16) ----
    const int   row = n0 + (lane & 15);
    const float4 tr = ((const float4*)traj)[row];
    float hx = tr.z, hy = tr.w;
    const float invn = 1.0f / (sqrtf(hx*hx + hy*hy) + 1e-8f);
    hx *= invn; hy *= invn;
    const int   agent = row / TT;
    const float L   = veh_att[2*agent + 0];
    const float Wd  = veh_att[2*agent + 1];
    const int   mp  = mapixes[agent];
    const float pen = 0.5f * sqrtf(L*L + Wd*Wd);
    const float cxs = tr.x * inv_dx;            // center pre-scaled to pixel units
    const float cys = tr.y * inv_dx;

    // ---- A matrices, ISA 16x4 f32 layout: VGPR0 = K0(lo)/K2(hi), VGPR1 = K1(lo)/K3(hi)
    const bool hi = (lane >= 16);
    v2f A;                                      // pre-scaled by inv_dx
    A.x = (hi ? (L * hy)  : (L * hx))  * inv_dx;
    A.y = (hi ? (Wd * hx) : (-Wd * hy)) * inv_dx;
    v2f A2;                                     // row = [L^2, W^2, 0, 0]
    A2.x = hi ? 0.0f : (L * L);
    A2.y = hi ? 0.0f : (Wd * Wd);

    // ---- per-D-slot row constants: slot r holds row M = r + 8*(lane>=16) ----
    v8f   Cx, Cy;
    float pen_s[8];
    int   base_s[8];                            // byte offset of map plane, < 64MB
    #pragma unroll
    for (int r = 0; r < 8; ++r) {
        const int src = r + (hi ? 8 : 0);
        Cx[r]     = __shfl(cxs, src, 32);
        Cy[r]     = __shfl(cys, src, 32);
        pen_s[r]  = __shfl(pen, src, 32);
        base_s[r] = __shfl(mp,  src, 32) * (RH * RW * 4);
    }

    int mind[8];                                // running min, bitcast f32
    #pragma unroll
    for (int r = 0; r < 8; ++r) mind[r] = __float_as_int(BIGF);

    const v8f czero = {};

    // ---- 13 point-tiles of 16 columns each ----
    for (int t = 0; t < NTILES; ++t) {
        int q = t * 16 + (lane & 15);
        q = (q < NP) ? q : (NP - 1);            // tail duplicates: min-safe
        const float uu = -0.5f + (float)(q / PVV) * (1.0f / 9.0f);
        const float vv = -0.5f + (float)(q % PVV) * (1.0f / 19.0f);

        // B matrices, 4x16: VGPR0 = K0(lo)/K2(hi), VGPR1 = K1(lo)/K3(hi)
        v2f Bx, By, Bd;
        Bx.x = hi ? 0.0f : uu;        Bx.y = hi ? 0.0f : vv;
        By.x = hi ? uu   : 0.0f;      By.y = hi ? vv   : 0.0f;
        Bd.x = hi ? 0.0f : uu * uu;   Bd.y = hi ? 0.0f : vv * vv;

        v8f Dx = __builtin_amdgcn_wmma_f32_16x16x4_f32(
            false, A,  false, Bx, (short)0, Cx,    false, false);   // px*inv_dx
        v8f Dy = __builtin_amdgcn_wmma_f32_16x16x4_f32(
            false, A,  false, By, (short)0, Cy,    false, false);   // py*inv_dx
        v8f Dd = __builtin_amdgcn_wmma_f32_16x16x4_f32(
            false, A2, false, Bd, (short)0, czero, false, false);   // d2

        #pragma unroll
        for (int r = 0; r < 8; ++r) {
            // trunc == floor+clip here: they only differ for negatives, which clamp to 0
            int ix = (int)Dx[r];
            int iy = (int)Dy[r];
            ix = min(max(ix, 0), RW - 1);
            iy = min(max(iy, 0), RH - 1);
            const unsigned voff = (unsigned)base_s[r]
                                + (((unsigned)(iy << 11) + (unsigned)ix) << 2);
            const float val = *(const float*)((const char*)raster + voff);  // GVS gather
            // raster val is 0 or 1: non-drivable(0) keeps d2, drivable(1) pushes to ~1e30
            const int d2i = __float_as_int(fmaf(val, BIGF, Dd[r]));
            mind[r] = min(mind[r], d2i);        // v_min_i32, valid for +ve floats
        }
    }

    // ---- reduce min over the 16 columns (lanes within each 16-lane half) ----
    #pragma unroll
    for (int r = 0; r < 8; ++r) {
        int m = mind[r];
        m = min(m, __shfl_xor(m, 1, 32));
        m = min(m, __shfl_xor(m, 2, 32));
        m = min(m, __shfl_xor(m, 4, 32));
        m = min(m, __shfl_xor(m, 8, 32));
        mind[r] = m;
    }

    // lane 0 writes rows n0..n0+7, lane 16 writes rows n0+8..n0+15
    if ((lane & 15) == 0) {
        const int rowbase = n0 + (hi ? 8 : 0);
        #pragma unroll
        for (int r = 0; r < 8; ++r) {
            const float m = __int_as_float(mind[r]);
            out[rowbase + r] = (m < COLLT) ? (1.0f - sqrtf(m) / pen_s[r]) : 0.0f;
        }
    }
}

extern "C" void kernel_launch(void* const* d_in, const int* in_sizes, int n_in,
                              void* d_out, int out_size, void* d_ws, size_t ws_size,
                              hipStream_t stream) {
    const float* traj    = (const float*)d_in[0];
    const float* veh_att = (const float*)d_in[1];
    const float* raster  = (const float*)d_in[2];
    const int*   mapixes = (const int*)d_in[3];
    const float* dx      = (const float*)d_in[4];
    float* out = (float*)d_out;

    const int waves = NTOT / 16;               // 1600 waves, 16 rows each
    const int wavesPerBlock = 4;               // 128 threads -> 400 blocks
    dim3 block(32 * wavesPerBlock);
    dim3 grid(waves / wavesPerBlock);
    env_coll_kernel<<<grid, block, 0, stream>>>(traj, veh_att, raster, mapixes, dx, out);
}